// SilLossROI_49933289783819
// MI455X (gfx1250) — compile-verified
//
#include <hip/hip_runtime.h>
#include <hip/hip_bf16.h>
#include <math.h>

typedef __attribute__((ext_vector_type(2))) float v2f;
typedef __attribute__((ext_vector_type(8))) float v8f;

#define BATCH 2
#define NV 192
#define NF 352
#define REND 256
#define IMG (BATCH * REND * REND)   // 131072
#define SIGMA 0.7f
#define NEG_LOG2E (-1.44269504088896340736f)

#if __has_builtin(__builtin_amdgcn_exp2f)
#define EXP2(x) __builtin_amdgcn_exp2f(x)   // raw v_exp_f32 (2^x)
#else
#define EXP2(x) exp2f(x)
#endif

// ---------------------------------------------------------------------------
// Stage 1: transform + project vertices, build per-edge affine coefficients.
// The -log2(e) factor of the sigmoid's exp is folded into the coefficients,
// so the WMMA output u = -log2(e) * (d*inv) and exp(-d*inv) = v_exp_f32(u).
// Stored in the WMMA B-matrix lane layout:
//   coefLo[(b*3+k)*NF + f] = (cA, cB)   -> read by lanes 0-15 (K=0,1)
//   coefHi[(b*3+k)*NF + f] = (cC, 0)    -> read by lanes 16-31 (K=2,3)
// ---------------------------------------------------------------------------
__global__ __launch_bounds__(1024) void setup_kernel(
    const float* __restrict__ R, const float* __restrict__ obj_t,
    const float* __restrict__ obj_s, const float* __restrict__ vertices,
    const int* __restrict__ faces, const float* __restrict__ K,
    float2* __restrict__ coefLo, float2* __restrict__ coefHi) {
  __shared__ float2 p2d[BATCH * NV];
  int t = threadIdx.x;
  if (t < BATCH * NV) {
    int b = t / NV, v = t % NV;
    const float* vert = vertices + (b * NV + v) * 3;
    const float* Rb = R + b * 9;
    const float* Kb = K + b * 9;
    float x = vert[0], y = vert[1], z = vert[2];
    float s = obj_s[b];
    // verts = s * (vertices @ R + obj_t)   (row-vec times matrix)
    float X = s * (x * Rb[0] + y * Rb[3] + z * Rb[6] + obj_t[b * 3 + 0]);
    float Y = s * (x * Rb[1] + y * Rb[4] + z * Rb[7] + obj_t[b * 3 + 1]);
    float Z = s * (x * Rb[2] + y * Rb[5] + z * Rb[8] + obj_t[b * 3 + 2]);
    float u = Kb[0] * X / Z + Kb[2];
    float w = Kb[4] * Y / Z + Kb[5];
    p2d[t] = make_float2(u * (float)REND, w * (float)REND);
  }
  __syncthreads();
  if (t < BATCH * NF) {
    int b = t / NF, f = t % NF;
    int i0 = faces[f * 3 + 0], i1 = faces[f * 3 + 1], i2 = faces[f * 3 + 2];
    float2 t0 = p2d[b * NV + i0];
    float2 t1 = p2d[b * NV + i1];
    float2 t2 = p2d[b * NV + i2];
    float e01x = t1.x - t0.x, e01y = t1.y - t0.y;
    float e02x = t2.x - t0.x, e02y = t2.y - t0.y;
    float area2 = e01x * e02y - e01y * e02x;
    float orient = (area2 >= 0.0f) ? 1.0f : -1.0f;
    float2 pts[3] = {t0, t1, t2};
#pragma unroll
    for (int k = 0; k < 3; ++k) {
      float2 pa = pts[k];
      float2 pb = pts[(k + 1) % 3];
      float ex = pb.x - pa.x, ey = pb.y - pa.y;
      // edge-function scale, with -log2(e) folded in for the exp2 path
      float inv = orient / ((sqrtf(ex * ex + ey * ey) + 1e-8f) * SIGMA) *
                  NEG_LOG2E;
      int idx = (b * 3 + k) * NF + f;
      coefLo[idx] = make_float2(-inv * ey, inv * ex);          // px, py coeffs
      coefHi[idx] = make_float2(inv * (ey * pa.x - ex * pa.y), // constant
                                0.0f);
    }
  }
}

// ---------------------------------------------------------------------------
// Stage 2: soft silhouette via V_WMMA_F32_16X16X4_F32
//   A (16x4): rows = 16 consecutive pixels of one image row; K = {px,py,1,0}
//   B (4x16): columns = 16 faces of one edge group; K = {cA,cB,cC,0}
//   D (16x16) = -log2(e)*d*inv for every (pixel, face) of the tile
// One wave handles 16 pixels; loops over 22 face tiles x 3 edge groups.
// Sigmoid product fused: s0*s1*s2 = 1 / ((1+e0)(1+e1)(1+e2)), single v_rcp;
// each e_i is a single v_exp_f32 of the WMMA output (no pre-scale).
// ---------------------------------------------------------------------------
__global__ __launch_bounds__(256) void sil_kernel(
    const float2* __restrict__ coefLo, const float2* __restrict__ coefHi,
    const float* __restrict__ keep_mask, float* __restrict__ out_image) {
  int wave = blockIdx.x * 8 + (threadIdx.x >> 5);
  int lane = threadIdx.x & 31;
  int x0 = (wave & 15) << 4;    // 16 tiles per row
  int y = (wave >> 4) & 255;    // image row
  int b = wave >> 12;           // batch
  int m = lane & 15;
  int half = lane >> 4;

  float px = (float)(x0 + m) + 0.5f;
  float py = (float)y + 0.5f;

  // A-matrix lane layout: lanes 0-15 hold K=0,1 ; lanes 16-31 hold K=2,3
  v2f amat;
  amat.x = half ? 1.0f : px;
  amat.y = half ? 0.0f : py;

  const float2* src = (half ? coefHi : coefLo) + b * 3 * NF;

  float acc[8];
#pragma unroll
  for (int r = 0; r < 8; ++r) acc[r] = 1.0f;

  for (int tile = 0; tile < NF / 16; ++tile) {
    int fcol = tile * 16 + m;
    float2 c0 = src[0 * NF + fcol];
    float2 c1 = src[1 * NF + fcol];
    float2 c2 = src[2 * NF + fcol];
    v2f b0; b0.x = c0.x; b0.y = c0.y;
    v2f b1; b1.x = c1.x; b1.y = c1.y;
    v2f b2; b2.x = c2.x; b2.y = c2.y;
    v8f cz = {};
    v8f u0 = __builtin_amdgcn_wmma_f32_16x16x4_f32(false, amat, false, b0,
                                                   (short)0, cz, false, false);
    v8f u1 = __builtin_amdgcn_wmma_f32_16x16x4_f32(false, amat, false, b1,
                                                   (short)0, cz, false, false);
    v8f u2 = __builtin_amdgcn_wmma_f32_16x16x4_f32(false, amat, false, b2,
                                                   (short)0, cz, false, false);
#pragma unroll
    for (int r = 0; r < 8; ++r) {
      float e0 = EXP2(u0[r]);   // = exp(-d0*inv0)
      float e1 = EXP2(u1[r]);
      float e2 = EXP2(u2[r]);
      // denom = (1+e0)(1+e1)(1+e2); p = sig0*sig1*sig2 = 1/denom
      float denom = (1.0f + e0) * (1.0f + e1) * (1.0f + e2);
      float p = __builtin_amdgcn_rcpf(denom);      // raw v_rcp_f32
      // clip(p, 0, 1-1e-4) then (1-p)  ==  max(1-p, 1e-4)
      acc[r] *= fmaxf(1.0f - p, 1e-4f);
    }
  }

  // product over the 16 face columns (lanes of each half-wave)
#pragma unroll
  for (int mask = 1; mask < 16; mask <<= 1) {
#pragma unroll
    for (int r = 0; r < 8; ++r)
      acc[r] *= __shfl_xor(acc[r], mask, 32);
  }

  // C/D layout: VGPR r <-> pixel x0 + r + 8*half
  if (m == 0) {
#pragma unroll
    for (int r = 0; r < 8; ++r) {
      int x = x0 + half * 8 + r;
      int idx = (b * REND + y) * REND + x;
      out_image[idx] = keep_mask[idx] * (1.0f - acc[r]);
    }
  }
}

// ---------------------------------------------------------------------------
// Stage 3: edges = maxpool7(image) - image ; copy passthroughs ;
//          per-block partial sums of (image - image_ref)^2
// ---------------------------------------------------------------------------
__global__ __launch_bounds__(256) void post_kernel(
    const float* __restrict__ image, const float* __restrict__ image_ref,
    const float* __restrict__ edt, float* __restrict__ d_out,
    float* __restrict__ partials) {
  int gid = blockIdx.x * 256 + threadIdx.x;  // 0 .. IMG-1
  int b = gid >> 16;
  int y = (gid >> 8) & 255;
  int x = gid & 255;

  float im = image[gid];
  const float* plane = image + b * REND * REND;
  int y0 = max(y - 3, 0), y1 = min(y + 3, REND - 1);
  int x0 = max(x - 3, 0), x1 = min(x + 3, REND - 1);
  float mp = -INFINITY;
  for (int yy = y0; yy <= y1; ++yy)
    for (int xx = x0; xx <= x1; ++xx)
      mp = fmaxf(mp, plane[yy * REND + xx]);

  d_out[1 + IMG + gid] = mp - im;          // edges
  float ref = image_ref[gid];
  d_out[1 + 2 * IMG + gid] = ref;          // image_ref passthrough
  d_out[1 + 3 * IMG + gid] = edt[gid];     // edt_ref_edge passthrough

  __shared__ float sh[256];
  float diff = im - ref;
  sh[threadIdx.x] = diff * diff;
  __syncthreads();
  for (int s = 128; s > 0; s >>= 1) {
    if (threadIdx.x < s) sh[threadIdx.x] += sh[threadIdx.x + s];
    __syncthreads();
  }
  if (threadIdx.x == 0) partials[blockIdx.x] = sh[0];
}

__global__ __launch_bounds__(512) void reduce_kernel(
    const float* __restrict__ partials, float* __restrict__ d_out) {
  __shared__ float sh[512];
  int t = threadIdx.x;
  sh[t] = partials[t];
  __syncthreads();
  for (int s = 256; s > 0; s >>= 1) {
    if (t < s) sh[t] += sh[t + s];
    __syncthreads();
  }
  if (t == 0) d_out[0] = sh[0] * (1.0f / (float)BATCH);  // mean over batch
}

// ---------------------------------------------------------------------------
extern "C" void kernel_launch(void* const* d_in, const int* in_sizes, int n_in,
                              void* d_out, int out_size, void* d_ws,
                              size_t ws_size, hipStream_t stream) {
  const float* R        = (const float*)d_in[0];
  const float* obj_t    = (const float*)d_in[1];
  const float* obj_s    = (const float*)d_in[2];
  const float* vertices = (const float*)d_in[3];
  const int*   faces    = (const int*)d_in[4];
  const float* K        = (const float*)d_in[5];
  const float* keep     = (const float*)d_in[6];
  const float* img_ref  = (const float*)d_in[7];
  const float* edt      = (const float*)d_in[8];
  float* out = (float*)d_out;

  float2* coefLo = (float2*)d_ws;                 // 2112 float2
  float2* coefHi = coefLo + BATCH * 3 * NF;       // 2112 float2
  float*  partials = (float*)(coefHi + BATCH * 3 * NF);  // 512 floats

  setup_kernel<<<1, 1024, 0, stream>>>(R, obj_t, obj_s, vertices, faces, K,
                                       coefLo, coefHi);
  // 8192 waves = BATCH*256 rows * 16 tiles; 8 waves per 256-thread block
  sil_kernel<<<1024, 256, 0, stream>>>(coefLo, coefHi, keep, out + 1);
  post_kernel<<<IMG / 256, 256, 0, stream>>>(out + 1, img_ref, edt, out,
                                             partials);
  reduce_kernel<<<1, 512, 0, stream>>>(partials, out);
}